// Attention_60498909331516
// MI455X (gfx1250) — compile-verified
//
#include <hip/hip_runtime.h>

typedef __attribute__((ext_vector_type(2))) float v2f;
typedef __attribute__((ext_vector_type(8))) float v8f;

constexpr int S_DIM = 2048;
constexpr int T_DIM = 2048;
constexpr int B_DIM = 8;
constexpr int H_DIM = 1024;

constexpr int BM  = 128;     // block tile M
constexpr int BN  = 128;     // block tile N
constexpr int BK  = 16;      // K chunk staged in LDS
constexpr int LDP = BK + 2;  // padded LDS row stride (floats): float2 stays 8B-aligned, banks spread

// Waves: 8 per block, tiled 2 (M) x 4 (N); each wave owns a 64x32 tile = 4x2 WMMA tiles.

// ---------------------------------------------------------------------------
// Phase 1: E[t,b,s] = sum_h dec[t,b,h] * enc[s,b,h]   -> Wbuf [T,B,S]
// ---------------------------------------------------------------------------
__global__ __launch_bounds__(256) void energies_wmma(
    const float* __restrict__ enc,   // [S,B,H]
    const float* __restrict__ dec,   // [T,B,H]
    float* __restrict__ Wbuf)        // [T,B,S]
{
  __shared__ float As[2][BM][LDP];   // dec tile [t_local][k]
  __shared__ float Bs[2][BN][LDP];   // enc tile [s_local][k]

  const int b  = blockIdx.z;
  const int t0 = blockIdx.x * BM;
  const int s0 = blockIdx.y * BN;

  const int tid  = threadIdx.x;
  const int lane = tid & 31;
  const int wave = tid >> 5;
  const int wmId = wave & 1;    // M group (2 x 64 rows)
  const int wnId = wave >> 1;   // N group (4 x 32 cols)
  const int ln   = lane & 15;
  const int kg   = lane >> 4;   // K pair group (0/1)

  // global->LDS copy mapping: 4 threads per 16-wide row, 64 rows per pass
  const int crow = tid >> 2;        // 0..63
  const int ccol = (tid & 3) * 4;   // 0,4,8,12

  v8f c[4][2] = {};
  float4 aR[2], bR[2];

  auto loadA = [&](int k0) {
#pragma unroll
    for (int i = 0; i < 2; ++i)
      aR[i] = *(const float4*)(dec + ((size_t)(t0 + crow + i * 64) * B_DIM + b) * H_DIM + k0 + ccol);
  };
  auto loadB = [&](int k0) {
#pragma unroll
    for (int i = 0; i < 2; ++i)
      bR[i] = *(const float4*)(enc + ((size_t)(s0 + crow + i * 64) * B_DIM + b) * H_DIM + k0 + ccol);
  };
  auto stage = [&](int buf) {
#pragma unroll
    for (int i = 0; i < 2; ++i) {
      const int row = crow + i * 64;
      *(float2*)&As[buf][row][ccol]     = make_float2(aR[i].x, aR[i].y);
      *(float2*)&As[buf][row][ccol + 2] = make_float2(aR[i].z, aR[i].w);
      *(float2*)&Bs[buf][row][ccol]     = make_float2(bR[i].x, bR[i].y);
      *(float2*)&Bs[buf][row][ccol + 2] = make_float2(bR[i].z, bR[i].w);
    }
  };

  constexpr int NS = H_DIM / BK;   // 64 stages
  loadA(0); loadB(0);
  stage(0);
  __syncthreads();

  for (int s = 0; s < NS; ++s) {
    const int buf = s & 1;
    if (s + 1 < NS) { loadA((s + 1) * BK); loadB((s + 1) * BK); }

#pragma unroll
    for (int kk = 0; kk < BK; kk += 4) {
      v2f a[4], bb[2];
#pragma unroll
      for (int i = 0; i < 4; ++i)
        a[i] = *(const v2f*)&As[buf][wmId * 64 + i * 16 + ln][kk + 2 * kg];
#pragma unroll
      for (int j = 0; j < 2; ++j)
        bb[j] = *(const v2f*)&Bs[buf][wnId * 32 + j * 16 + ln][kk + 2 * kg];
#pragma unroll
      for (int i = 0; i < 4; ++i)
#pragma unroll
        for (int j = 0; j < 2; ++j)
          c[i][j] = __builtin_amdgcn_wmma_f32_16x16x4_f32(
              false, a[i], false, bb[j], (short)0, c[i][j], false, false);
    }

    if (s + 1 < NS) stage(buf ^ 1);
    __syncthreads();
  }

#pragma unroll
  for (int i = 0; i < 4; ++i) {
#pragma unroll
    for (int j = 0; j < 2; ++j) {
      const int n_g = s0 + wnId * 32 + j * 16 + ln;
#pragma unroll
      for (int r = 0; r < 8; ++r) {
        const int m_g = t0 + wmId * 64 + i * 16 + kg * 8 + r;
        Wbuf[((size_t)m_g * B_DIM + b) * S_DIM + n_g] = c[i][j][r];
      }
    }
  }
}

// ---------------------------------------------------------------------------
// Phase 2: softmax over last dim of Wbuf [T,B,S], in place. One block per row.
// ---------------------------------------------------------------------------
__global__ __launch_bounds__(256) void softmax_rows(float* __restrict__ Wbuf)
{
  __shared__ float red[256];
  float* p = Wbuf + (size_t)blockIdx.x * S_DIM;
  const int tid = threadIdx.x;

  float4 x0 = *(float4*)(p + tid * 8);
  float4 x1 = *(float4*)(p + tid * 8 + 4);

  float m = fmaxf(fmaxf(fmaxf(x0.x, x0.y), fmaxf(x0.z, x0.w)),
                  fmaxf(fmaxf(x1.x, x1.y), fmaxf(x1.z, x1.w)));
  red[tid] = m;
  __syncthreads();
  for (int off = 128; off > 0; off >>= 1) {
    if (tid < off) red[tid] = fmaxf(red[tid], red[tid + off]);
    __syncthreads();
  }
  m = red[0];
  __syncthreads();

  float4 e0, e1;
  e0.x = expf(x0.x - m); e0.y = expf(x0.y - m);
  e0.z = expf(x0.z - m); e0.w = expf(x0.w - m);
  e1.x = expf(x1.x - m); e1.y = expf(x1.y - m);
  e1.z = expf(x1.z - m); e1.w = expf(x1.w - m);

  red[tid] = (e0.x + e0.y + e0.z + e0.w) + (e1.x + e1.y + e1.z + e1.w);
  __syncthreads();
  for (int off = 128; off > 0; off >>= 1) {
    if (tid < off) red[tid] = red[tid] + red[tid + off];
    __syncthreads();
  }
  const float inv = 1.0f / red[0];

  e0.x *= inv; e0.y *= inv; e0.z *= inv; e0.w *= inv;
  e1.x *= inv; e1.y *= inv; e1.z *= inv; e1.w *= inv;
  *(float4*)(p + tid * 8)     = e0;
  *(float4*)(p + tid * 8 + 4) = e1;
}

// ---------------------------------------------------------------------------
// Phase 3: ctx[t,b,h] = sum_s W[t,b,s] * enc[s,b,h]
// ---------------------------------------------------------------------------
__global__ __launch_bounds__(256) void context_wmma(
    const float* __restrict__ enc,   // [S,B,H]
    const float* __restrict__ W,     // [T,B,S] (softmaxed)
    float* __restrict__ ctx)         // [T,B,H]
{
  __shared__ float As[2][BM][LDP];   // W tile            [t_local][k=s_local]
  __shared__ float Bs[2][BN][LDP];   // enc tile (transp) [n=h_local][k=s_local]

  const int b  = blockIdx.z;
  const int t0 = blockIdx.x * BM;
  const int h0 = blockIdx.y * BN;

  const int tid  = threadIdx.x;
  const int lane = tid & 31;
  const int wave = tid >> 5;
  const int wmId = wave & 1;
  const int wnId = wave >> 1;
  const int ln   = lane & 15;
  const int kg   = lane >> 4;

  const int crow  = tid >> 2;        // A copy: 0..63
  const int ccol  = (tid & 3) * 4;
  const int srow8 = tid >> 5;        // B copy: 0..7 (k rows), 32 threads per 128-wide row
  const int hcol  = (tid & 31) * 4;

  v8f c[4][2] = {};
  float4 aR[2], bR[2];

  auto loadA = [&](int k0) {
#pragma unroll
    for (int i = 0; i < 2; ++i)
      aR[i] = *(const float4*)(W + ((size_t)(t0 + crow + i * 64) * B_DIM + b) * S_DIM + k0 + ccol);
  };
  auto loadB = [&](int k0) {
#pragma unroll
    for (int i = 0; i < 2; ++i)
      bR[i] = *(const float4*)(enc + ((size_t)(k0 + srow8 + i * 8) * B_DIM + b) * H_DIM + h0 + hcol);
  };
  auto stage = [&](int buf) {
#pragma unroll
    for (int i = 0; i < 2; ++i) {
      const int row = crow + i * 64;
      *(float2*)&As[buf][row][ccol]     = make_float2(aR[i].x, aR[i].y);
      *(float2*)&As[buf][row][ccol + 2] = make_float2(aR[i].z, aR[i].w);
      const int sr = srow8 + i * 8;    // transpose enc into [h][s]
      Bs[buf][hcol + 0][sr] = bR[i].x;
      Bs[buf][hcol + 1][sr] = bR[i].y;
      Bs[buf][hcol + 2][sr] = bR[i].z;
      Bs[buf][hcol + 3][sr] = bR[i].w;
    }
  };

  constexpr int NS = S_DIM / BK;   // 128 stages
  loadA(0); loadB(0);
  stage(0);
  __syncthreads();

  for (int s = 0; s < NS; ++s) {
    const int buf = s & 1;
    if (s + 1 < NS) { loadA((s + 1) * BK); loadB((s + 1) * BK); }

#pragma unroll
    for (int kk = 0; kk < BK; kk += 4) {
      v2f a[4], bb[2];
#pragma unroll
      for (int i = 0; i < 4; ++i)
        a[i] = *(const v2f*)&As[buf][wmId * 64 + i * 16 + ln][kk + 2 * kg];
#pragma unroll
      for (int j = 0; j < 2; ++j)
        bb[j] = *(const v2f*)&Bs[buf][wnId * 32 + j * 16 + ln][kk + 2 * kg];
#pragma unroll
      for (int i = 0; i < 4; ++i)
#pragma unroll
        for (int j = 0; j < 2; ++j)
          c[i][j] = __builtin_amdgcn_wmma_f32_16x16x4_f32(
              false, a[i], false, bb[j], (short)0, c[i][j], false, false);
    }

    if (s + 1 < NS) stage(buf ^ 1);
    __syncthreads();
  }

#pragma unroll
  for (int i = 0; i < 4; ++i) {
#pragma unroll
    for (int j = 0; j < 2; ++j) {
      const int n_g = h0 + wnId * 32 + j * 16 + ln;
#pragma unroll
      for (int r = 0; r < 8; ++r) {
        const int m_g = t0 + wmId * 64 + i * 16 + kg * 8 + r;
        ctx[((size_t)m_g * B_DIM + b) * H_DIM + n_g] = c[i][j][r];
      }
    }
  }
}

// ---------------------------------------------------------------------------
extern "C" void kernel_launch(void* const* d_in, const int* in_sizes, int n_in,
                              void* d_out, int out_size, void* d_ws, size_t ws_size,
                              hipStream_t stream) {
  const float* enc = (const float*)d_in[0];   // encoder_output [S,B,H]
  const float* dec = (const float*)d_in[1];   // decoder_hidden [T,B,H]
  float* ctx  = (float*)d_out;                                    // [T,B,H]
  float* Wbuf = (float*)d_out + (size_t)T_DIM * B_DIM * H_DIM;    // [T,B,S]

  dim3 blk(256);
  energies_wmma<<<dim3(T_DIM / BM, S_DIM / BN, B_DIM), blk, 0, stream>>>(enc, dec, Wbuf);
  softmax_rows<<<dim3(T_DIM * B_DIM), blk, 0, stream>>>(Wbuf);
  context_wmma<<<dim3(T_DIM / BM, H_DIM / BN, B_DIM), blk, 0, stream>>>(enc, Wbuf, ctx);
}